// EdgeAutoEncoderMultiTask_67130338836695
// MI455X (gfx1250) — compile-verified
//
#include <hip/hip_runtime.h>

#define EPSF 1e-5f

typedef __attribute__((ext_vector_type(2))) float v2f;
typedef __attribute__((ext_vector_type(8))) float v8f;

// ---------------------------------------------------------------------------
// utility kernels
// ---------------------------------------------------------------------------
__global__ void fill_kernel(float* __restrict__ p, float val, long long count) {
  long long i = (long long)blockIdx.x * blockDim.x + threadIdx.x;
  long long stride = (long long)gridDim.x * blockDim.x;
  for (; i < count; i += stride) p[i] = val;
}

__global__ void deg_scatter_kernel(const int* __restrict__ dst, float* __restrict__ deg, int E) {
  int e = blockIdx.x * blockDim.x + threadIdx.x;
  if (e < E) atomicAdd(&deg[dst[e]], 1.0f);
}

__global__ void deg_rsqrt_kernel(float* __restrict__ d, int n) {
  int i = blockIdx.x * blockDim.x + threadIdx.x;
  if (i < n) d[i] = rsqrtf(d[i]);
}

// pad rows: dst[Kdst x OUT] = src[Ksrc x OUT] with zero rows for k >= Ksrc
__global__ void pad_rows_kernel(const float* __restrict__ src, float* __restrict__ dst,
                                int Ksrc, int Kdst, int OUT) {
  int tid = blockIdx.x * blockDim.x + threadIdx.x;
  if (tid >= Kdst * OUT) return;
  int k = tid / OUT;
  dst[tid] = (k < Ksrc) ? src[k * OUT + (tid % OUT)] : 0.0f;
}

// xp[i*4+k] = (k<3) ? x[i*3+k] : 0
__global__ void pad_x_kernel(const float* __restrict__ x, float* __restrict__ xp, int n) {
  long long tid = (long long)blockIdx.x * blockDim.x + threadIdx.x;
  if (tid >= (long long)n * 4) return;
  int k = (int)(tid & 3);
  long long i = tid >> 2;
  xp[tid] = (k < 3) ? x[i * 3 + k] : 0.0f;
}

// ---------------------------------------------------------------------------
// WMMA f32 GEMM: Y[rows x OUT] = X[rows x K] @ W[K x OUT] (+bias) (+relu)
// One wave handles a 16-row strip, V_WMMA_F32_16X16X4_F32, K % 4 == 0.
// Weights staged in LDS once per block (shared by all 8 waves); B fragments
// come from ds_load, A fragments are branch-free float2 global loads.
//   A frag (16x4 f32, 2 VGPR): lanes 0-15 hold K=0,1; lanes 16-31 hold K=2,3
//   B frag (4x16 f32, 2 VGPR): same split, N striped across lanes
//   C/D (16x16 f32, 8 VGPR):   vgpr v, lane L -> C[v + 8*(L>=16)][L%16]
// DUAL: second weight matrix sharing the same A fragments (enc vs res path).
// ---------------------------------------------------------------------------
template <int K, int OUT, bool RELU, bool DUAL>
__global__ __launch_bounds__(256) void wmma_gemm_kernel(
    const float* __restrict__ X, int nrows,
    const float* __restrict__ W0, const float* __restrict__ B0, float* __restrict__ Y0,
    const float* __restrict__ W1, const float* __restrict__ B1, float* __restrict__ Y1) {
  static_assert(K % 4 == 0, "K must be a multiple of 4");
  constexpr int NT  = OUT / 16;
  constexpr int WSZ = K * OUT;
  __shared__ float sW[(DUAL ? 2 : 1) * WSZ];

  // cooperative weight staging (before any early-out: barrier must be uniform)
  for (int idx = threadIdx.x; idx < WSZ; idx += 256) {
    sW[idx] = W0[idx];
    if (DUAL) sW[WSZ + idx] = W1[idx];
  }
  __syncthreads();

  int tid  = blockIdx.x * 256 + threadIdx.x;
  int wid  = tid >> 5;
  int lane = tid & 31;
  int row0 = wid * 16;
  if (row0 >= nrows) return;
  int g = lane >> 4;   // lane-group: selects K pair / M+8 half
  int m = lane & 15;   // row (A) / col (B, C)

  const float* xrow = X + (long long)(row0 + m) * K + 2 * g;

  v8f zeroc = {};
  v8f acc0[NT];
  v8f acc1[DUAL ? NT : 1];
#pragma unroll
  for (int t = 0; t < NT; ++t) {
    acc0[t] = zeroc;
    if (DUAL) acc1[t] = zeroc;
  }

#pragma unroll
  for (int kk = 0; kk < K; kk += 4) {
    int ka = kk + 2 * g;
    v2f a = *(const v2f*)(xrow + kk);  // 8B-aligned: K%4==0, 2g even
#pragma unroll
    for (int t = 0; t < NT; ++t) {
      v2f b0;
      b0.x = sW[(ka + 0) * OUT + t * 16 + m];
      b0.y = sW[(ka + 1) * OUT + t * 16 + m];
      acc0[t] = __builtin_amdgcn_wmma_f32_16x16x4_f32(
          false, a, false, b0, (short)0, acc0[t], false, false);
      if (DUAL) {
        v2f b1;
        b1.x = sW[WSZ + (ka + 0) * OUT + t * 16 + m];
        b1.y = sW[WSZ + (ka + 1) * OUT + t * 16 + m];
        acc1[t] = __builtin_amdgcn_wmma_f32_16x16x4_f32(
            false, a, false, b1, (short)0, acc1[t], false, false);
      }
    }
  }

#pragma unroll
  for (int t = 0; t < NT; ++t) {
    float bias0 = B0 ? B0[t * 16 + m] : 0.0f;
    float bias1 = (DUAL && B1) ? B1[t * 16 + m] : 0.0f;
#pragma unroll
    for (int v = 0; v < 8; ++v) {
      long long row = row0 + v + 8 * g;
      float val = acc0[t][v] + bias0;
      if (RELU) val = fmaxf(val, 0.0f);
      Y0[row * OUT + t * 16 + m] = val;
      if (DUAL) Y1[row * OUT + t * 16 + m] = acc1[t][v] + bias1;
    }
  }
}

// ---------------------------------------------------------------------------
// edge scatter: agg[dst] += h[src] * dis[src]*dis[dst]
// ---------------------------------------------------------------------------
__global__ void edge_scatter64_kernel(const int* __restrict__ src, const int* __restrict__ dst,
                                      const float* __restrict__ dis, const float* __restrict__ h,
                                      float* __restrict__ agg, int E) {
  long long tid = (long long)blockIdx.x * blockDim.x + threadIdx.x;
  if (tid >= (long long)E * 16) return;
  int e = (int)(tid >> 4);
  int q = (int)(tid & 15);
  int s = src[e], d = dst[e];
  float c = dis[s] * dis[d];
  float4 hv = *((const float4*)(h + (long long)s * 64) + q);
  float* base = agg + (long long)d * 64 + q * 4;
  atomicAdd(base + 0, hv.x * c);
  atomicAdd(base + 1, hv.y * c);
  atomicAdd(base + 2, hv.z * c);
  atomicAdd(base + 3, hv.w * c);
}

__global__ void edge_scatter32_kernel(const int* __restrict__ src, const int* __restrict__ dst,
                                      const float* __restrict__ dis, const float* __restrict__ h,
                                      float* __restrict__ agg, int E) {
  long long tid = (long long)blockIdx.x * blockDim.x + threadIdx.x;
  if (tid >= (long long)E * 8) return;
  int e = (int)(tid >> 3);
  int q = (int)(tid & 7);
  int s = src[e], d = dst[e];
  float c = dis[s] * dis[d];
  float4 hv = *((const float4*)(h + (long long)s * 32) + q);
  float* base = agg + (long long)d * 32 + q * 4;
  atomicAdd(base + 0, hv.x * c);
  atomicAdd(base + 1, hv.y * c);
  atomicAdd(base + 2, hv.z * c);
  atomicAdd(base + 3, hv.w * c);
}

__global__ void edge_scatter1_kernel(const int* __restrict__ src, const int* __restrict__ dst,
                                     const float* __restrict__ dis, const float* __restrict__ h,
                                     float* __restrict__ agg, int E) {
  int e = blockIdx.x * blockDim.x + threadIdx.x;
  if (e >= E) return;
  int s = src[e], d = dst[e];
  atomicAdd(&agg[d], h[s] * dis[s] * dis[d]);
}

// ---------------------------------------------------------------------------
// combine: out = relu(BN(agg + h*dis^2 + conv_b)) + res     (in place on res)
// ---------------------------------------------------------------------------
template <int F>
__global__ void combine_kernel(const float* __restrict__ agg, const float* __restrict__ h,
                               const float* __restrict__ dis, const float* __restrict__ conv_b,
                               const float* __restrict__ bn_g, const float* __restrict__ bn_b,
                               const float* __restrict__ bn_m, const float* __restrict__ bn_v,
                               float* __restrict__ res_io, int n) {
  long long tid = (long long)blockIdx.x * blockDim.x + threadIdx.x;
  if (tid >= (long long)n * F) return;
  int i = (int)(tid / F);
  int f = (int)(tid % F);
  float s = dis[i];
  float val = agg[tid] + h[tid] * s * s + conv_b[f];
  val = (val - bn_m[f]) * (bn_g[f] * rsqrtf(bn_v[f] + EPSF)) + bn_b[f];
  val = fmaxf(val, 0.0f);
  res_io[tid] = val + res_io[tid];
}

// bottleneck feature transform: hz[i] = dot(h[i,0:32], w[0:32])
__global__ void dotz_kernel(const float* __restrict__ h, const float* __restrict__ w,
                            float* __restrict__ hz, int n) {
  int i = blockIdx.x * blockDim.x + threadIdx.x;
  if (i >= n) return;
  const float* row = h + (long long)i * 32;
  float acc = 0.0f;
#pragma unroll
  for (int f = 0; f < 32; ++f) acc = fmaf(row[f], w[f], acc);
  hz[i] = acc;
}

// z = aggz + hz*dis^2 + b; emit z output, time head, and d1 = relu(z*dec1_w+b)
__global__ void zfinal_kernel(const float* __restrict__ aggz, const float* __restrict__ hz,
                              const float* __restrict__ dis, const float* __restrict__ enc3_b,
                              const float* __restrict__ th1_w, const float* __restrict__ th1_b,
                              const float* __restrict__ th2_w, const float* __restrict__ th2_b,
                              const float* __restrict__ dec1_w, const float* __restrict__ dec1_b,
                              float* __restrict__ out_z, float* __restrict__ out_t,
                              float* __restrict__ d1, int n) {
  int i = blockIdx.x * blockDim.x + threadIdx.x;
  if (i >= n) return;
  float s = dis[i];
  float z = aggz[i] + hz[i] * s * s + enc3_b[0];
  out_z[i] = z;
  float tacc = th2_b[0];
#pragma unroll
  for (int j = 0; j < 16; ++j)
    tacc = fmaf(fmaxf(fmaf(z, th1_w[j], th1_b[j]), 0.0f), th2_w[j], tacc);
  out_t[i] = tacc;
  float* d1r = d1 + (long long)i * 32;
#pragma unroll
  for (int j = 0; j < 32; ++j)
    d1r[j] = fmaxf(fmaf(z, dec1_w[j], dec1_b[j]), 0.0f);
}

// recon[i,c] = sum_k d2[i,k]*dec3_w[k,c] + dec3_b[c]
__global__ void dec3_kernel(const float* __restrict__ d2, const float* __restrict__ w,
                            const float* __restrict__ b, float* __restrict__ recon, int n) {
  int i = blockIdx.x * blockDim.x + threadIdx.x;
  if (i >= n) return;
  const float* row = d2 + (long long)i * 64;
  float a0 = b[0], a1 = b[1], a2 = b[2];
#pragma unroll
  for (int k = 0; k < 64; ++k) {
    float v = row[k];
    a0 = fmaf(v, w[k * 3 + 0], a0);
    a1 = fmaf(v, w[k * 3 + 1], a1);
    a2 = fmaf(v, w[k * 3 + 2], a2);
  }
  float* o = recon + (long long)i * 3;
  o[0] = a0; o[1] = a1; o[2] = a2;
}

// ---------------------------------------------------------------------------
extern "C" void kernel_launch(void* const* d_in, const int* in_sizes, int n_in,
                              void* d_out, int out_size, void* d_ws, size_t ws_size,
                              hipStream_t stream) {
  const float* x   = (const float*)d_in[0];
  const int*   ei  = (const int*)d_in[1];
  const int N = in_sizes[0] / 3;
  const int E = in_sizes[1] / 2;
  const int* src = ei;
  const int* dst = ei + E;

  const float* enc1_w = (const float*)d_in[2];
  const float* enc1_b = (const float*)d_in[3];
  const float* res1_w = (const float*)d_in[4];
  const float* res1_b = (const float*)d_in[5];
  const float* bn1_g  = (const float*)d_in[6];
  const float* bn1_b  = (const float*)d_in[7];
  const float* bn1_m  = (const float*)d_in[8];
  const float* bn1_v  = (const float*)d_in[9];
  const float* enc2_w = (const float*)d_in[10];
  const float* enc2_b = (const float*)d_in[11];
  const float* res2_w = (const float*)d_in[12];
  const float* res2_b = (const float*)d_in[13];
  const float* bn2_g  = (const float*)d_in[14];
  const float* bn2_b  = (const float*)d_in[15];
  const float* bn2_m  = (const float*)d_in[16];
  const float* bn2_v  = (const float*)d_in[17];
  const float* enc3_w = (const float*)d_in[18];
  const float* enc3_b = (const float*)d_in[19];
  const float* dec1_w = (const float*)d_in[20];
  const float* dec1_b = (const float*)d_in[21];
  const float* dec2_w = (const float*)d_in[22];
  const float* dec2_b = (const float*)d_in[23];
  const float* dec3_w = (const float*)d_in[24];
  const float* dec3_b = (const float*)d_in[25];
  const float* th1_w  = (const float*)d_in[26];
  const float* th1_b  = (const float*)d_in[27];
  const float* th2_w  = (const float*)d_in[28];
  const float* th2_b  = (const float*)d_in[29];

  // workspace layout (floats): dis | bufA(64N) | bufB(64N) | bufC(64N)
  float* ws   = (float*)d_ws;
  float* dis  = ws;
  float* bufA = ws + (long long)N;          // h1 -> agg2 -> d1
  float* bufB = ws + 65LL * N;              // res1/h1o -> hz,aggz
  float* bufC = ws + 129LL * N;             // xp,padded W -> agg1 -> h2x,res2/h2o -> d2

  float* h1   = bufA;
  float* res1 = bufB;                       // becomes h1o in place
  float* xp   = bufC;                       // padded x [N x 4], dead after gemm1
  float* w1p  = bufC + 4LL * N;             // padded enc1_w [4 x 64]
  float* w2p  = w1p + 256;                  // padded res1_w [4 x 64]
  float* agg1 = bufC;
  float* h2x  = bufC;
  float* res2 = bufC + 32LL * N;            // becomes h2o in place
  float* agg2 = bufA;
  float* hz   = bufB;
  float* aggz = bufB + N;
  float* d1   = bufA;
  float* d2   = bufC;

  float* out_recon = (float*)d_out;
  float* out_t     = out_recon + 3LL * N;
  float* out_z     = out_recon + 4LL * N;

  const int B = 256;
  auto blocks = [B](long long work) { return (unsigned)((work + B - 1) / B); };
  const long long gemm_threads = ((long long)(N + 15) / 16) * 32;

  // 1) degrees -> dis = rsqrt(1 + indeg)
  fill_kernel<<<blocks(N), B, 0, stream>>>(dis, 1.0f, N);
  deg_scatter_kernel<<<blocks(E), B, 0, stream>>>(dst, dis, E);
  deg_rsqrt_kernel<<<blocks(N), B, 0, stream>>>(dis, N);

  // 2) block 1: pad K=3 -> 4, then h1 = xp@enc1_wp ; res1 = xp@res1_wp + b
  pad_x_kernel<<<blocks(4LL * N), B, 0, stream>>>(x, xp, N);
  pad_rows_kernel<<<blocks(256), B, 0, stream>>>(enc1_w, w1p, 3, 4, 64);
  pad_rows_kernel<<<blocks(256), B, 0, stream>>>(res1_w, w2p, 3, 4, 64);
  wmma_gemm_kernel<4, 64, false, true><<<blocks(gemm_threads), B, 0, stream>>>(
      xp, N, w1p, nullptr, h1, w2p, res1_b, res1);
  fill_kernel<<<blocks(64LL * N), B, 0, stream>>>(agg1, 0.0f, 64LL * N);
  edge_scatter64_kernel<<<blocks(16LL * E), B, 0, stream>>>(src, dst, dis, h1, agg1, E);
  combine_kernel<64><<<blocks(64LL * N), B, 0, stream>>>(
      agg1, h1, dis, enc1_b, bn1_g, bn1_b, bn1_m, bn1_v, res1, N);  // res1 -> h1o

  // 3) block 2: h2x = h1o@enc2_w ; res2 = h1o@res2_w + b
  wmma_gemm_kernel<64, 32, false, true><<<blocks(gemm_threads), B, 0, stream>>>(
      res1, N, enc2_w, nullptr, h2x, res2_w, res2_b, res2);
  fill_kernel<<<blocks(32LL * N), B, 0, stream>>>(agg2, 0.0f, 32LL * N);
  edge_scatter32_kernel<<<blocks(8LL * E), B, 0, stream>>>(src, dst, dis, h2x, agg2, E);
  combine_kernel<32><<<blocks(32LL * N), B, 0, stream>>>(
      agg2, h2x, dis, enc2_b, bn2_g, bn2_b, bn2_m, bn2_v, res2, N);  // res2 -> h2o

  // 4) bottleneck z
  dotz_kernel<<<blocks(N), B, 0, stream>>>(res2, enc3_w, hz, N);
  fill_kernel<<<blocks(N), B, 0, stream>>>(aggz, 0.0f, N);
  edge_scatter1_kernel<<<blocks(E), B, 0, stream>>>(src, dst, dis, hz, aggz, E);
  zfinal_kernel<<<blocks(N), B, 0, stream>>>(aggz, hz, dis, enc3_b,
                                             th1_w, th1_b, th2_w, th2_b,
                                             dec1_w, dec1_b, out_z, out_t, d1, N);

  // 5) decoder: d2 = relu(d1@dec2_w + b) via WMMA, then recon
  wmma_gemm_kernel<32, 64, true, false><<<blocks(gemm_threads), B, 0, stream>>>(
      d1, N, dec2_w, dec2_b, d2, nullptr, nullptr, nullptr);
  dec3_kernel<<<blocks(N), B, 0, stream>>>(d2, dec3_w, dec3_b, out_recon, N);

  (void)n_in; (void)out_size; (void)ws_size;
}